// StructuredReadout_47287589929655
// MI455X (gfx1250) — compile-verified
//
#include <hip/hip_runtime.h>

// Row gather: out[r, :] = node_states[idx[r], :]
// R = 100,000 rows, D = 256 fp32 (1 KB/row). Pure HBM-bandwidth bound
// (~205 MB total -> ~8.8 us at 23.3 TB/s). Optimal CDNA5 mix: wave32
// coalesced global_load_b128 / global_store_b128, scalarized index load,
// SALU address math.

typedef __attribute__((ext_vector_type(4))) float f4;

// Fast path: D == 256 floats (64 f4 per row), one wave32 per row.
// Lane l moves f4 elements l and l+32 -> the wave streams the whole 1 KB
// row as 64 coalesced b128 transactions.
__global__ __launch_bounds__(256) void StructuredReadout_gather_d256(
    const float* __restrict__ nodes,   // [N, 256] fp32
    const int*   __restrict__ idx,     // [R] int32 row indices
    float*       __restrict__ out,     // [R, 256] fp32
    int R) {
  const int lane = threadIdx.x & 31;
  // Force wave-in-block id into an SGPR so idx[] is loaded once per wave
  // (scalar path) and row-base address math happens on the SALU.
  const int waveInBlock = __builtin_amdgcn_readfirstlane((int)(threadIdx.x >> 5));
  const int row = blockIdx.x * 8 + waveInBlock;
  if (row >= R) return;

  const int srcRow = idx[row];  // uniform -> s_load

  const f4* __restrict__ src = (const f4*)nodes + ((long long)srcRow << 6);
  f4*       __restrict__ dst = (f4*)out        + ((long long)row    << 6);

  // Issue both loads before the stores: two outstanding b128 loads per lane,
  // single loadcnt wait.
  f4 a = src[lane];
  f4 b = src[lane + 32];
  __builtin_nontemporal_store(a, dst + lane);        // write-once output:
  __builtin_nontemporal_store(b, dst + lane + 32);   // don't pollute L2
}

// Generic fallback for arbitrary D (multiple of 4 floats).
__global__ __launch_bounds__(256) void StructuredReadout_gather_generic(
    const float* __restrict__ nodes,
    const int*   __restrict__ idx,
    float*       __restrict__ out,
    int R, int D) {
  const int lane = threadIdx.x & 31;
  const int waveInBlock = __builtin_amdgcn_readfirstlane((int)(threadIdx.x >> 5));
  const int row = blockIdx.x * (blockDim.x >> 5) + waveInBlock;
  if (row >= R) return;

  const int srcRow = idx[row];
  const f4* __restrict__ src = (const f4*)(nodes + (long long)srcRow * D);
  f4*       __restrict__ dst = (f4*)(out + (long long)row * D);
  const int nvec = D >> 2;
  for (int v = lane; v < nvec; v += 32) {
    f4 x = src[v];
    __builtin_nontemporal_store(x, dst + v);
  }
}

extern "C" void kernel_launch(void* const* d_in, const int* in_sizes, int n_in,
                              void* d_out, int out_size, void* d_ws, size_t ws_size,
                              hipStream_t stream) {
  const float* nodes = (const float*)d_in[0];   // node_states [N, D]
  const int*   idx   = (const int*)d_in[1];     // readout_indices [R]
  float*       out   = (float*)d_out;           // [R, D]

  const int R = in_sizes[1];
  if (R <= 0) return;
  const int D = out_size / R;                   // 256 for this problem

  const int threads = 256;                      // 8 waves -> 8 rows per block
  const int blocks = (R + 7) / 8;

  if (D == 256) {
    StructuredReadout_gather_d256<<<blocks, threads, 0, stream>>>(
        nodes, idx, out, R);
  } else if (D > 0) {
    StructuredReadout_gather_generic<<<blocks, threads, 0, stream>>>(
        nodes, idx, out, R, D);
  }
}